// PointCloudChamferLoss_79070347920144
// MI455X (gfx1250) — compile-verified
//
#include <hip/hip_runtime.h>

#define BB 16
#define VV 100000
#define SS 4096
#define TILES_PER_BATCH (SS / 16)              // 256
#define TILES_TOTAL (BB * TILES_PER_BATCH)     // 4096

typedef float v2f __attribute__((ext_vector_type(2)));
typedef float v8f __attribute__((ext_vector_type(8)));

// ---------------------------------------------------------------------------
// 1) Gather: resolve int64 indices into packed float4 (x, y, z, |p|^2)
// ---------------------------------------------------------------------------
__global__ void chamfer_gather_kernel(const float* __restrict__ src_verts,
                                      const float* __restrict__ dst_verts,
                                      const long long* __restrict__ src_idx,
                                      const long long* __restrict__ dst_idx,
                                      float4* __restrict__ Psrc,
                                      float4* __restrict__ Pdst) {
    int t = blockIdx.x * blockDim.x + threadIdx.x;
    const int n = BB * SS;
    if (t >= 2 * n) return;
    int cloud = t / n;          // 0 = src, 1 = dst
    int j = t - cloud * n;      // (b, i) flattened
    int b = j / SS;
    const float* verts = cloud ? dst_verts : src_verts;
    const long long* idx = cloud ? dst_idx : src_idx;
    long long id = idx[j];
    const float* p = verts + ((long long)b * VV + id) * 3;
    float x = p[0], y = p[1], z = p[2];
    float4 o = make_float4(x, y, z, x * x + y * y + z * z);
    if (cloud) Pdst[j] = o; else Psrc[j] = o;
}

// ---------------------------------------------------------------------------
// 2) One wave owns 16 "row" points of one batch; scans all 4096 "column"
//    points in 16-wide tiles via V_WMMA_F32_16X16X4_F32.
//    dist(m,n) = A_m . B_n + d2_n, A=[x,y,z,s2], B=[-2u,-2v,-2w,1].
// ---------------------------------------------------------------------------
__global__ void chamfer_pass_kernel(const float4* __restrict__ Psrc,
                                    const float4* __restrict__ Pdst,
                                    float* __restrict__ partials) {
    int wave = (blockIdx.x * blockDim.x + threadIdx.x) >> 5;
    int lane = threadIdx.x & 31;
    int dir  = wave / TILES_TOTAL;              // 0: rows=src cols=dst, 1: swapped
    int tile = wave - dir * TILES_TOTAL;
    int b    = tile / TILES_PER_BATCH;
    int m0   = (tile % TILES_PER_BATCH) * 16;

    const float4* rows = dir ? Pdst : Psrc;
    const float4* cols = dir ? Psrc : Pdst;
    rows += (size_t)b * SS;
    cols += (size_t)b * SS;

    const int  l15 = lane & 15;
    const bool hi  = lane >= 16;

    // A operand: 16x4, lanes 0-15 hold {K0,K1}={x,y}, lanes 16-31 hold {K2,K3}={z,s2}
    float4 p = rows[m0 + l15];
    v2f a;
    a.x = hi ? p.z : p.x;
    a.y = hi ? p.w : p.y;

    float rowmin[8];
#pragma unroll
    for (int r = 0; r < 8; ++r) rowmin[r] = 3.402823466e38f;

    for (int n0 = 0; n0 < SS; n0 += 16) {
        // Both lane halves load the SAME column point (column = n0 + l15),
        // so q.w == |d|^2 is available in every lane.
        float4 q = cols[n0 + l15];
        v2f bv;
        bv.x = -2.0f * (hi ? q.z : q.x);        // K0 = -2u (lo) / K2 = -2w (hi)
        bv.y = hi ? 1.0f : (-2.0f * q.y);       // K1 = -2v (lo) / K3 = 1   (hi)
        float d2 = q.w;

        v8f c = {};
        c = __builtin_amdgcn_wmma_f32_16x16x4_f32(
                /*neg_a=*/false, a, /*neg_b=*/false, bv,
                /*c_mod=*/(short)0, c, /*reuse_a=*/false, /*reuse_b=*/false);

#pragma unroll
        for (int r = 0; r < 8; ++r) {
            float v = fmaxf(c[r] + d2, 0.0f);   // clamp like reference
            rowmin[r] = fminf(rowmin[r], v);
        }
    }

    // Per-row min across the 16 lanes of each half, then sum the 16 row minima.
    float sum = 0.0f;
#pragma unroll
    for (int r = 0; r < 8; ++r) {
        float rm = rowmin[r];
        rm = fminf(rm, __shfl_xor(rm, 1, 32));
        rm = fminf(rm, __shfl_xor(rm, 2, 32));
        rm = fminf(rm, __shfl_xor(rm, 4, 32));
        rm = fminf(rm, __shfl_xor(rm, 8, 32));
        sum += rm;                              // lanes 0-15: rows m0+r ; 16-31: m0+8+r
    }
    sum += __shfl_xor(sum, 16, 32);             // combine halves -> all 16 rows
    if (lane == 0) partials[wave] = sum;
}

// ---------------------------------------------------------------------------
// 3) Deterministic final reduction of 8192 partials -> scalar mean
// ---------------------------------------------------------------------------
__global__ void chamfer_reduce_kernel(const float* __restrict__ partials,
                                      float* __restrict__ out) {
    __shared__ float sm[256];
    float s = 0.0f;
    for (int i = threadIdx.x; i < 2 * TILES_TOTAL; i += 256) s += partials[i];
    sm[threadIdx.x] = s;
    __syncthreads();
    for (int off = 128; off > 0; off >>= 1) {
        if ((int)threadIdx.x < off) sm[threadIdx.x] += sm[threadIdx.x + off];
        __syncthreads();
    }
    if (threadIdx.x == 0) out[0] = sm[0] * (1.0f / (float)(BB * SS));
}

extern "C" void kernel_launch(void* const* d_in, const int* in_sizes, int n_in,
                              void* d_out, int out_size, void* d_ws, size_t ws_size,
                              hipStream_t stream) {
    const float* src_verts     = (const float*)d_in[0];      // (B, V, 3) f32
    const float* dst_verts     = (const float*)d_in[1];      // (B, V, 3) f32
    const long long* src_idx   = (const long long*)d_in[2];  // (B, S) i64
    const long long* dst_idx   = (const long long*)d_in[3];  // (B, S) i64
    float* out = (float*)d_out;

    char* ws = (char*)d_ws;
    const size_t ptsBytes = (size_t)BB * SS * sizeof(float4);   // 1 MB each
    float4* Psrc    = (float4*)ws;
    float4* Pdst    = (float4*)(ws + ptsBytes);
    float*  partial = (float*)(ws + 2 * ptsBytes);              // 8192 floats

    int nGather = 2 * BB * SS;
    chamfer_gather_kernel<<<(nGather + 255) / 256, 256, 0, stream>>>(
        src_verts, dst_verts, src_idx, dst_idx, Psrc, Pdst);

    int waves = 2 * TILES_TOTAL;   // 8192 waves, 8 waves per 256-thread block
    chamfer_pass_kernel<<<waves / 8, 256, 0, stream>>>(Psrc, Pdst, partial);

    chamfer_reduce_kernel<<<1, 256, 0, stream>>>(partial, out);
}